// MambaLayer_10960756539726
// MI455X (gfx1250) — compile-verified
//
#include <hip/hip_runtime.h>
#include <hip/hip_bf16.h>
#include <math.h>

// ---------------------------------------------------------------------------
// VMamba SS2D block for MI455X (gfx1250, wave32).
// GEMMs use V_WMMA_F32_16X16X4_F32 (exact fp32, matches reference precision).
// Scan kernel stages B/C tiles with GLOBAL_LOAD_ASYNC_TO_LDS_B32 (ASYNCcnt).
// ---------------------------------------------------------------------------

typedef float v2f __attribute__((ext_vector_type(2)));
typedef float v8f __attribute__((ext_vector_type(8)));

constexpr int BB = 2;
constexpr int HH = 48;
constexpr int WW = 48;
constexpr int LL = HH * WW;          // 2304
constexpr int BL = BB * LL;          // 4608
constexpr int DM = 128;              // d_model
constexpr int DI = 256;              // d_inner
constexpr int KD = 4;                // scan directions
constexpr int RR = 8;                // dt_rank
constexpr int NS = 16;               // d_state
constexpr int CD = RR + 2 * NS;      // 40 rows of x_dbl

__device__ __forceinline__ v8f wmma4(v2f a, v2f b, v8f c) {
    // D = A(16x4,f32) * B(4x16,f32) + C(16x16,f32)
    return __builtin_amdgcn_wmma_f32_16x16x4_f32(
        /*neg_a=*/false, a, /*neg_b=*/false, b,
        /*c_mod=*/(short)0, c, /*reuse_a=*/false, /*reuse_b=*/false);
}

__device__ __forceinline__ float silu_f(float x) {
    return x / (1.0f + __expf(-x));
}

// Async copy of one dword from global memory directly into LDS (no VGPR
// data path); tracked by ASYNCcnt.
__device__ __forceinline__ void async_g2lds_b32(unsigned lds_off,
                                                unsigned long long gaddr) {
    asm volatile("global_load_async_to_lds_b32 %0, %1, off"
                 :: "v"(lds_off), "v"(gaddr) : "memory");
}
__device__ __forceinline__ void wait_asynccnt0() {
    asm volatile("s_wait_asynccnt 0x0" ::: "memory");
}

// ---------------------------------------------------------------------------
// Kernel 1: xz[BL,512] = X[BL,128] @ in_proj_w[512,128]^T   (WMMA fp32)
// One 16x16 tile per wave; K-loop over 128 in steps of 4.
// ---------------------------------------------------------------------------
__global__ __launch_bounds__(256) void k_inproj(const float* __restrict__ X,
                                                const float* __restrict__ Wp,
                                                float* __restrict__ xz) {
    const int wave = blockIdx.x * 8 + (threadIdx.x >> 5);
    const int lane = threadIdx.x & 31;
    const int MT = BL / 16;                 // 288 row tiles
    const int tm = wave % MT;
    const int tn = wave / MT;               // < 32 col tiles
    const int m0 = tm * 16, n0 = tn * 16;
    const int r = lane & 15;
    const int koff = (lane >> 4) << 1;      // half-wave holds K+0/K+1 or K+2/K+3

    const float* Arow = X  + (size_t)(m0 + r) * DM;
    const float* Brow = Wp + (size_t)(n0 + r) * DM;   // B[k][n] = Wp[n][k]
    __builtin_prefetch(Arow, 0, 0);
    __builtin_prefetch(Brow, 0, 0);

    v8f acc = {};
    for (int k0 = 0; k0 < DM; k0 += 4) {
        v2f a, b;
        a[0] = Arow[k0 + koff];
        a[1] = Arow[k0 + koff + 1];
        b[0] = Brow[k0 + koff];
        b[1] = Brow[k0 + koff + 1];
        acc = wmma4(a, b, acc);
    }
    const int rs = m0 + ((lane >> 4) << 3);  // M = i (lanes 0-15) / i+8 (16-31)
    const int cc = n0 + r;
#pragma unroll
    for (int i = 0; i < 8; ++i)
        xz[(size_t)(rs + i) * (2 * DI) + cc] = acc[i];
}

// ---------------------------------------------------------------------------
// Kernel 2: depthwise 3x3 conv (pad 1) + bias + SiLU, scatter into the four
// scan orderings xs[b,k,d,l].
// ---------------------------------------------------------------------------
__global__ __launch_bounds__(256) void k_conv(const float* __restrict__ xz,
                                              const float* __restrict__ cw,
                                              const float* __restrict__ cb,
                                              float* __restrict__ xs) {
    const int idx = blockIdx.x * 256 + threadIdx.x;   // (b,l,d), d fastest
    const int d = idx & 255;
    const int t = idx >> 8;
    const int l = t % LL;
    const int b = t / LL;
    const int h = l / WW, w = l % WW;

    float s = cb[d];
#pragma unroll
    for (int kh = 0; kh < 3; ++kh) {
        const int h2 = h + kh - 1;
        if (h2 < 0 || h2 >= HH) continue;
#pragma unroll
        for (int kw = 0; kw < 3; ++kw) {
            const int w2 = w + kw - 1;
            if (w2 < 0 || w2 >= WW) continue;
            s += xz[(size_t)((b * LL + h2 * WW + w2)) * (2 * DI) + d] *
                 cw[d * 9 + kh * 3 + kw];
        }
    }
    s = silu_f(s);

    const int lwh = w * HH + h;
    xs[((size_t)(b * KD + 0) * DI + d) * LL + l]            = s;
    xs[((size_t)(b * KD + 1) * DI + d) * LL + lwh]          = s;
    xs[((size_t)(b * KD + 2) * DI + d) * LL + (LL - 1 - l)] = s;
    xs[((size_t)(b * KD + 3) * DI + d) * LL + (LL - 1 - lwh)] = s;
}

// ---------------------------------------------------------------------------
// Kernel 3: x_dbl[bk, c, l] = x_proj_w[k, c, :] . xs[bk, :, l]   (WMMA fp32)
// M=40 padded to 48 via zero-masked A rows; K=256, N=2304.
// ---------------------------------------------------------------------------
__global__ __launch_bounds__(256) void k_xdbl(const float* __restrict__ xpw,
                                              const float* __restrict__ xs,
                                              float* __restrict__ xdbl) {
    const int wave = blockIdx.x * 8 + (threadIdx.x >> 5);
    const int lane = threadIdx.x & 31;
    const int TPK = 3 * (LL / 16);          // 432 tiles per (b,k)
    const int bk = wave / TPK;
    const int t = wave % TPK;
    const int tm = t % 3, tn = t / 3;
    const int k = bk & 3;
    const int m0 = tm * 16, n0 = tn * 16;
    const int r = lane & 15;
    const int koff = (lane >> 4) << 1;

    const int crow = m0 + r;
    const float amask = (crow < CD) ? 1.0f : 0.0f;
    const float* Arow = xpw + ((size_t)k * CD + (crow < CD ? crow : 0)) * DI;
    const float* Bcol = xs + (size_t)bk * DI * LL + (n0 + r);
    __builtin_prefetch(Arow, 0, 0);

    v8f acc = {};
    for (int k0 = 0; k0 < DI; k0 += 4) {
        v2f a, b;
        a[0] = Arow[k0 + koff]     * amask;
        a[1] = Arow[k0 + koff + 1] * amask;
        b[0] = Bcol[(size_t)(k0 + koff)     * LL];
        b[1] = Bcol[(size_t)(k0 + koff + 1) * LL];
        acc = wmma4(a, b, acc);
    }
    const int rs = m0 + ((lane >> 4) << 3);
    const int cc = n0 + r;
#pragma unroll
    for (int i = 0; i < 8; ++i) {
        const int row = rs + i;
        if (row < CD)
            xdbl[((size_t)bk * CD + row) * LL + cc] = acc[i];
    }
}

// ---------------------------------------------------------------------------
// Kernel 4: delta[bk,d,l] = softplus( dts[bk,:,l] . dt_w[k,d,:] + dt_b[k,d] )
// ---------------------------------------------------------------------------
__global__ __launch_bounds__(256) void k_delta(const float* __restrict__ xdbl,
                                               const float* __restrict__ dtw,
                                               const float* __restrict__ dtb,
                                               float* __restrict__ delta) {
    const size_t idx = (size_t)blockIdx.x * 256 + threadIdx.x;  // (bk,d,l)
    const int l = (int)(idx % LL);
    const size_t t = idx / LL;
    const int d = (int)(t % DI);
    const int bk = (int)(t / DI);
    const int k = bk & 3;

    float s = dtb[k * DI + d];
#pragma unroll
    for (int r = 0; r < RR; ++r)
        s += xdbl[((size_t)bk * CD + r) * LL + l] *
             dtw[((size_t)k * DI + d) * RR + r];
    const float sp = (s > 20.0f) ? s : log1pf(__expf(s));
    delta[((size_t)bk * DI + d) * LL + l] = sp;
}

// ---------------------------------------------------------------------------
// Kernel 5: selective scan.  One block per (b,k); one thread per channel d.
// h[n] <- h[n]*exp(delta*A[n]) + delta*u*B[n,l];  y = sum_n h[n]*C[n,l] + D*u
// B/C (shared across channels) staged into LDS via async global->LDS copies
// (no VGPR data path, tracked by ASYNCcnt) in 64-step chunks.
// ---------------------------------------------------------------------------
__global__ __launch_bounds__(256) void k_scan(const float* __restrict__ delta,
                                              const float* __restrict__ xs,
                                              const float* __restrict__ xdbl,
                                              const float* __restrict__ Alog,
                                              const float* __restrict__ Dvs,
                                              float* __restrict__ ys) {
    __shared__ float sB[64][NS + 1];   // +1 pad to avoid bank conflicts
    __shared__ float sC[64][NS + 1];

    const int bk = blockIdx.x;
    const int k = bk & 3;
    const int d = threadIdx.x;

    float aN[NS];
#pragma unroll
    for (int n = 0; n < NS; ++n)
        aN[n] = -__expf(Alog[((size_t)k * DI + d) * NS + n]);
    const float Dv = Dvs[k * DI + d];

    const float* dl_p = delta + ((size_t)bk * DI + d) * LL;
    const float* u_p  = xs    + ((size_t)bk * DI + d) * LL;
    const float* Bp   = xdbl  + ((size_t)bk * CD + RR) * LL;       // rows 8..23
    const float* Cp   = xdbl  + ((size_t)bk * CD + RR + NS) * LL;  // rows 24..39
    float* y_p        = ys    + ((size_t)bk * DI + d) * LL;

    float hst[NS];
#pragma unroll
    for (int n = 0; n < NS; ++n) hst[n] = 0.0f;

    for (int l0 = 0; l0 < LL; l0 += 64) {
        // async stage: 64 l-steps x 16 states, coalesced on l; data flows
        // memory -> LDS without passing through the register file
        for (int i = threadIdx.x; i < 64 * NS; i += 256) {
            const int n = i >> 6;
            const int ll = i & 63;
            async_g2lds_b32((unsigned)(uintptr_t)&sB[ll][n],
                            (unsigned long long)(uintptr_t)(Bp + (size_t)n * LL + (l0 + ll)));
            async_g2lds_b32((unsigned)(uintptr_t)&sC[ll][n],
                            (unsigned long long)(uintptr_t)(Cp + (size_t)n * LL + (l0 + ll)));
        }
        wait_asynccnt0();   // this wave's async copies have landed in LDS
        __syncthreads();    // ... and every other wave's too

        for (int ll = 0; ll < 64; ++ll) {
            const int l = l0 + ll;
            const float dl = dl_p[l];
            const float u = u_p[l];
            const float du = dl * u;
            float y = 0.0f;
#pragma unroll
            for (int n = 0; n < NS; ++n) {
                hst[n] = hst[n] * __expf(dl * aN[n]) + du * sB[ll][n];
                y += hst[n] * sC[ll][n];
            }
            y_p[l] = y + Dv * u;
        }
        __syncthreads();
    }
}

// ---------------------------------------------------------------------------
// Kernel 6a: merge 4 directions + LayerNorm + SiLU(z) gate.
// Writes the gated result over the (now dead) xh half of xz.
// ---------------------------------------------------------------------------
__global__ __launch_bounds__(256) void k_combine(const float* __restrict__ ys,
                                                 const float* __restrict__ lng,
                                                 const float* __restrict__ lnb,
                                                 float* __restrict__ xz) {
    __shared__ float red[256];
    const int bl = blockIdx.x;
    const int b = bl / LL;
    const int l = bl % LL;
    const int d = threadIdx.x;
    const int h = l / WW, w = l % WW;
    const int lwh = w * HH + h;

    const float v =
        ys[((size_t)(b * KD + 0) * DI + d) * LL + l] +
        ys[((size_t)(b * KD + 2) * DI + d) * LL + (LL - 1 - l)] +
        ys[((size_t)(b * KD + 1) * DI + d) * LL + lwh] +
        ys[((size_t)(b * KD + 3) * DI + d) * LL + (LL - 1 - lwh)];

    red[d] = v;
    __syncthreads();
    for (int s = 128; s > 0; s >>= 1) {
        if (d < s) red[d] += red[d + s];
        __syncthreads();
    }
    const float mu = red[0] * (1.0f / DI);
    __syncthreads();

    const float dv = v - mu;
    red[d] = dv * dv;
    __syncthreads();
    for (int s = 128; s > 0; s >>= 1) {
        if (d < s) red[d] += red[d + s];
        __syncthreads();
    }
    const float var = red[0] * (1.0f / DI);

    const float yn = dv * rsqrtf(var + 1e-5f) * lng[d] + lnb[d];
    const float z = xz[(size_t)bl * (2 * DI) + DI + d];
    xz[(size_t)bl * (2 * DI) + d] = yn * silu_f(z);
}

// ---------------------------------------------------------------------------
// Kernel 6b: out[BL,128] = yg[BL,256] @ out_w[128,256]^T   (WMMA fp32)
// yg lives in xz rows with stride 512 (cols 0..255).
// ---------------------------------------------------------------------------
__global__ __launch_bounds__(256) void k_outproj(const float* __restrict__ xz,
                                                 const float* __restrict__ ow,
                                                 float* __restrict__ out) {
    const int wave = blockIdx.x * 8 + (threadIdx.x >> 5);
    const int lane = threadIdx.x & 31;
    const int MT = BL / 16;                 // 288 row tiles
    const int tm = wave % MT;
    const int tn = wave / MT;               // < 8 col tiles
    const int m0 = tm * 16, n0 = tn * 16;
    const int r = lane & 15;
    const int koff = (lane >> 4) << 1;

    const float* Arow = xz + (size_t)(m0 + r) * (2 * DI);
    const float* Brow = ow + (size_t)(n0 + r) * DI;
    __builtin_prefetch(Arow, 0, 0);
    __builtin_prefetch(Brow, 0, 0);

    v8f acc = {};
    for (int k0 = 0; k0 < DI; k0 += 4) {
        v2f a, b;
        a[0] = Arow[k0 + koff];
        a[1] = Arow[k0 + koff + 1];
        b[0] = Brow[k0 + koff];
        b[1] = Brow[k0 + koff + 1];
        acc = wmma4(a, b, acc);
    }
    const int rs = m0 + ((lane >> 4) << 3);
    const int cc = n0 + r;
#pragma unroll
    for (int i = 0; i < 8; ++i)
        out[(size_t)(rs + i) * DM + cc] = acc[i];
}

// ---------------------------------------------------------------------------
// Host launcher
// ---------------------------------------------------------------------------
extern "C" void kernel_launch(void* const* d_in, const int* in_sizes, int n_in,
                              void* d_out, int out_size, void* d_ws, size_t ws_size,
                              hipStream_t stream) {
    const float* x    = (const float*)d_in[0];
    const float* ipw  = (const float*)d_in[1];
    const float* cw   = (const float*)d_in[2];
    const float* cb   = (const float*)d_in[3];
    const float* xpw  = (const float*)d_in[4];
    const float* dtw  = (const float*)d_in[5];
    const float* dtb  = (const float*)d_in[6];
    const float* alog = (const float*)d_in[7];
    const float* dvs  = (const float*)d_in[8];
    const float* lng  = (const float*)d_in[9];
    const float* lnb  = (const float*)d_in[10];
    const float* ow   = (const float*)d_in[11];
    float* out = (float*)d_out;

    char* ws = (char*)d_ws;
    const size_t xz_bytes    = (size_t)BL * (2 * DI) * sizeof(float);       // 9.44 MB
    const size_t xs_bytes    = (size_t)BB * KD * DI * LL * sizeof(float);   // 18.9 MB
    const size_t xdbl_bytes  = (size_t)BB * KD * CD * LL * sizeof(float);   // 2.95 MB
    const size_t delta_bytes = xs_bytes;
    float* xz    = (float*)(ws);
    float* xs    = (float*)(ws + xz_bytes);
    float* xdbl  = (float*)(ws + xz_bytes + xs_bytes);
    float* delta = (float*)(ws + xz_bytes + xs_bytes + xdbl_bytes);
    float* ysb   = (float*)(ws + xz_bytes + xs_bytes + xdbl_bytes + delta_bytes);

    // 1) input projection: 288*32 = 9216 wave-tiles, 8 waves/block
    k_inproj<<<dim3(9216 / 8), dim3(256), 0, stream>>>(x, ipw, xz);

    // 2) depthwise conv + SiLU + 4-way scan scatter
    k_conv<<<dim3((BB * LL * DI) / 256), dim3(256), 0, stream>>>(xz, cw, cb, xs);

    // 3) x_dbl projection: 8 (b,k) * 432 tiles = 3456 waves
    k_xdbl<<<dim3(3456 / 8), dim3(256), 0, stream>>>(xpw, xs, xdbl);

    // 4) delta = softplus(...)
    k_delta<<<dim3((BB * KD * DI * LL) / 256), dim3(256), 0, stream>>>(xdbl, dtw, dtb, delta);

    // 5) selective scan: one block per (b,k)
    k_scan<<<dim3(BB * KD), dim3(256), 0, stream>>>(delta, xs, xdbl, alog, dvs, ysb);

    // 6a) merge + LayerNorm + gate
    k_combine<<<dim3(BL), dim3(256), 0, stream>>>(ysb, lng, lnb, xz);

    // 6b) output projection: 288*8 = 2304 wave-tiles
    k_outproj<<<dim3(2304 / 8), dim3(256), 0, stream>>>(xz, ow, out);
}